// RPN_67405216743952
// MI455X (gfx1250) — compile-verified
//
#include <hip/hip_runtime.h>
#include <math.h>

// ---------------- types ----------------
typedef __attribute__((ext_vector_type(2))) float v2f;
typedef __attribute__((ext_vector_type(8))) float v8f;

#define WMMA_F32(a, b, c) \
  __builtin_amdgcn_wmma_f32_16x16x4_f32(false, (a), false, (b), (short)0, (c), false, false)

// ---------------- problem constants ----------------
static constexpr int H = 128, W = 128, C = 512;   // feature map
static constexpr int HW = H * W;                  // 16384
static constexpr int K9 = 9 * C;                  // 4608 (3x3 conv K)
static constexpr int NCAT = 112;                  // 72+9+24 padded to 7*16
static constexpr int ATOT = HW * 9;               // 147456 anchors
static constexpr float IMG = 2048.0f;
static constexpr float NEGF = -3.4028234663852886e38f;

// ---------------- workspace layout (float units) ----------------
static constexpr size_t OFF_X      = 0;                         // 16384*512
static constexpr size_t OFF_WCAT   = OFF_X + (size_t)HW * C;    // 512*112
static constexpr size_t OFF_BCAT   = OFF_WCAT + 512 * NCAT;     // 112 (pad 128)
static constexpr size_t OFF_DELTAS = OFF_BCAT + 128;            // 16384*72
static constexpr size_t OFF_SCORES = OFF_DELTAS + (size_t)HW * 72; // 16384*9
static constexpr size_t OFF_ANCH   = OFF_SCORES + (size_t)HW * 9;  // 147456*4
static constexpr size_t OFF_BOXES  = OFF_ANCH + (size_t)ATOT * 4;  // 147456*4
static constexpr size_t OFF_NSC    = OFF_BOXES + (size_t)ATOT * 4; // 147456
static constexpr size_t OFF_FLAGS  = OFF_NSC + ATOT;            // 147456 ints
static constexpr size_t OFF_IDX    = OFF_FLAGS + ATOT;          // 147456 ints
static constexpr size_t OFF_CNT    = OFF_IDX + ATOT;            // 16

// =====================================================================
// Kernel 0: pack reg/cls/lm 1x1 weights into one [512,112] matrix + bias
// =====================================================================
__global__ void pack_weights(const float* __restrict__ regw, const float* __restrict__ clsw,
                             const float* __restrict__ lmw,  const float* __restrict__ regb,
                             const float* __restrict__ clsb, const float* __restrict__ lmb,
                             float* __restrict__ Wc, float* __restrict__ bc) {
  int t = blockIdx.x * blockDim.x + threadIdx.x;
  if (t < 512 * NCAT) {
    int k = t / NCAT, n = t % NCAT;
    float v = 0.0f;
    if (n < 72)       v = regw[k * 72 + n];
    else if (n < 81)  v = clsw[k * 9 + (n - 72)];
    else if (n < 105) v = lmw[k * 24 + (n - 81)];
    Wc[t] = v;
  }
  if (t < NCAT) {
    float v = 0.0f;
    if (t < 72)       v = regb[t];
    else if (t < 81)  v = clsb[t - 72];
    else if (t < 105) v = lmb[t - 81];
    bc[t] = v;
  }
}

// =====================================================================
// Kernel 1: 3x3 conv (SAME) + bias + ReLU as implicit GEMM via
// V_WMMA_F32_16X16X4_F32.  Wave tile = 64M x 64N (4x4 accumulators),
// workgroup = 8 waves = 128M x 256N.  Grid = 128 (M) x 2 (N) blocks.
// =====================================================================
__global__ __launch_bounds__(256) void conv3x3_wmma(
    const float* __restrict__ F,   // [128,128,512] NHWC
    const float* __restrict__ W1,  // [3,3,512,512] HWIO -> flat [4608,512]
    const float* __restrict__ B1,  // [512]
    float* __restrict__ X)         // [16384,512]
{
  const int lane = threadIdx.x & 31;
  const int wave = threadIdx.x >> 5;
  const int bm   = blockIdx.x & 127;
  const int bn   = blockIdx.x >> 7;
  const int mW   = bm * 128 + (wave & 1) * 64;       // wave M origin
  const int nW   = bn * 256 + (wave >> 1) * 64;      // wave N origin
  const int lm   = lane & 15;                        // A row / B,C col
  const int lk   = (lane >> 4) << 1;                 // k offset: 0 or 2

  v8f acc[4][4] = {};

  // per-mt spatial position (fixed per lane)
  int py[4], px[4];
  for (int mt = 0; mt < 4; ++mt) {
    int p = mW + mt * 16 + lm;
    py[mt] = p >> 7;
    px[mt] = p & 127;
  }

  for (int dydx = 0; dydx < 9; ++dydx) {
    const int dy = dydx / 3 - 1, dx = dydx % 3 - 1;
    const float* aB[4];
    float amask[4];
    for (int mt = 0; mt < 4; ++mt) {
      int iy = py[mt] + dy, ix = px[mt] + dx;
      bool ok = (iy >= 0) && (iy < H) && (ix >= 0) && (ix < W);
      int iyc = min(max(iy, 0), H - 1);
      int ixc = min(max(ix, 0), W - 1);
      aB[mt] = F + ((((size_t)iyc << 7) | (size_t)ixc) << 9) + lk;
      amask[mt] = ok ? 1.0f : 0.0f;
    }
    const float* wB = W1 + ((size_t)dydx * C + lk) * C;  // B[k][n] base for this dydx

    for (int c0 = 0; c0 < C; c0 += 4) {
      v2f a[4];
      for (int mt = 0; mt < 4; ++mt) {
        v2f t = *(const v2f*)(aB[mt] + c0);
        a[mt] = t * amask[mt];                     // zero-pad SAME borders
      }
      v2f b[4];
      const float* wp = wB + (size_t)c0 * C + nW + lm;
      for (int nt = 0; nt < 4; ++nt) {
        b[nt].x = wp[nt * 16];
        b[nt].y = wp[nt * 16 + C];
      }
      for (int mt = 0; mt < 4; ++mt)
        for (int nt = 0; nt < 4; ++nt)
          acc[mt][nt] = WMMA_F32(a[mt], b[nt], acc[mt][nt]);
    }
  }

  // store: C/D layout row = v + 8*(lane/16), col = lane%16
  const int rbase = (lane >> 4) << 3;
  for (int nt = 0; nt < 4; ++nt) {
    int n = nW + nt * 16 + lm;
    float bias = B1[n];
    for (int mt = 0; mt < 4; ++mt) {
      float* o = X + (size_t)(mW + mt * 16 + rbase) * C + n;
      v8f c = acc[mt][nt];
      for (int v = 0; v < 8; ++v) {
        float val = c[v] + bias;
        o[(size_t)v * C] = val > 0.0f ? val : 0.0f;
      }
    }
  }
}

// =====================================================================
// Kernel 2: fused 1x1 convs as GEMM [16384,512] x [512,112] via WMMA.
// Wave tile = 32M x 112N (2x7 accumulators), workgroup = 256M. 64 blocks.
// =====================================================================
__global__ __launch_bounds__(256) void gemm_heads(
    const float* __restrict__ X,    // [16384,512]
    const float* __restrict__ Wc,   // [512,112]
    const float* __restrict__ bc,   // [112]
    float* __restrict__ deltas,     // [16384,72]
    float* __restrict__ scoresRaw,  // [16384,9]
    float* __restrict__ lmOut)      // [16384,24]
{
  const int lane = threadIdx.x & 31;
  const int wave = threadIdx.x >> 5;
  const int mW   = blockIdx.x * 256 + wave * 32;
  const int lm   = lane & 15;
  const int lk   = (lane >> 4) << 1;

  v8f acc[2][7] = {};

  for (int k0 = 0; k0 < C; k0 += 4) {
    v2f a[2];
    for (int mt = 0; mt < 2; ++mt) {
      int p = mW + mt * 16 + lm;
      a[mt] = *(const v2f*)(X + (size_t)p * C + k0 + lk);
    }
    const float* wp = Wc + (size_t)(k0 + lk) * NCAT + lm;
    v2f b[7];
    for (int nt = 0; nt < 7; ++nt) {
      b[nt].x = wp[nt * 16];
      b[nt].y = wp[nt * 16 + NCAT];
    }
    for (int mt = 0; mt < 2; ++mt)
      for (int nt = 0; nt < 7; ++nt)
        acc[mt][nt] = WMMA_F32(a[mt], b[nt], acc[mt][nt]);
  }

  const int rbase = (lane >> 4) << 3;
  for (int nt = 0; nt < 7; ++nt) {
    int n = nt * 16 + lm;
    float bias = bc[n];
    for (int mt = 0; mt < 2; ++mt) {
      int mrow = mW + mt * 16 + rbase;
      v8f c = acc[mt][nt];
      for (int v = 0; v < 8; ++v) {
        int m = mrow + v;
        float val = c[v] + bias;
        if (n < 72)       deltas[(size_t)m * 72 + n] = val;
        else if (n < 81)  scoresRaw[(size_t)m * 9 + (n - 72)] = val;
        else if (n < 105) lmOut[(size_t)m * 24 + (n - 81)] = val;
      }
    }
  }
}

// =====================================================================
// Kernel 3: anchor generation (matches numpy fp32-sqrt then fp64 math)
// =====================================================================
__global__ void gen_anchors(float4* __restrict__ anchorsAll, int* __restrict__ flags) {
  int i = blockIdx.x * blockDim.x + threadIdx.x;
  if (i >= ATOT) return;
  int p = i / 9, t = i % 9;
  int y = p >> 7, x = p & 127;
  float cx = 8.0f + 16.0f * (float)x;
  float cy = 8.0f + 16.0f * (float)y;
  float s = (float)(8 << (t % 3));          // 8,16,32
  float r = 0.5f * (float)(1 << (t / 3));   // 0.5,1,2
  float hs = sqrtf((s * s) / r);            // fp32 like np.sqrt on f32
  double h = (double)hs * 16.0;             // * ys (fp64)
  double w = h * (double)r;                 // widths = heights*r*xs/ys, xs==ys
  float x0 = (float)((double)cx - w * 0.5);
  float y0 = (float)((double)cy - h * 0.5);
  float x1 = (float)((double)cx + w * 0.5);
  float y1 = (float)((double)cy + h * 0.5);
  anchorsAll[i] = make_float4(x0, y0, x1, y1);
  flags[i] = (x0 >= 0.0f) && (y0 >= 0.0f) && (x1 <= IMG) && (y1 <= IMG) ? 1 : 0;
}

// =====================================================================
// Kernel 4: stable stream compaction of inside anchors (LDS scan)
// =====================================================================
__global__ __launch_bounds__(1024) void compact_anchors(
    const float4* __restrict__ anchorsAll, const int* __restrict__ flags,
    int* __restrict__ idx, float4* __restrict__ outAnchors, int* __restrict__ counter) {
  __shared__ int sd[1024];
  __shared__ int base;
  const int tid = threadIdx.x;
  if (tid == 0) base = 0;
  __syncthreads();
  for (int chunk = 0; chunk < ATOT; chunk += 1024) {
    int i = chunk + tid;
    int f = flags[i];
    sd[tid] = f;
    __syncthreads();
    for (int off = 1; off < 1024; off <<= 1) {
      int v = (tid >= off) ? sd[tid - off] : 0;
      __syncthreads();
      sd[tid] += v;
      __syncthreads();
    }
    if (f) {
      int pos = base + sd[tid] - 1;   // exclusive position
      idx[pos] = i;
      outAnchors[pos] = anchorsAll[i];
    }
    __syncthreads();
    if (tid == 0) base += sd[1023];
    __syncthreads();
  }
  if (tid == 0) counter[0] = base;
}

// =====================================================================
// Kernel 5: box decode + clip + sigmoid(score)
// =====================================================================
__global__ void decode_boxes(const float4* __restrict__ anchorsAll, const int* __restrict__ idx,
                             const int* __restrict__ counter, const float* __restrict__ deltas,
                             const float* __restrict__ scoresRaw,
                             float4* __restrict__ boxes, float* __restrict__ nsc) {
  int j = blockIdx.x * blockDim.x + threadIdx.x;
  if (j >= counter[0]) return;
  int a = idx[j];
  int pos = a / 9, t = a % 9;
  const float* d = deltas + (size_t)pos * 72 + t * 8;
  float4 an = anchorsAll[a];
  float wa = an.z - an.x, ha = an.w - an.y;
  float xa = an.x + wa * 0.5f, ya = an.y + ha * 0.5f;
  float x1 = xa + wa * d[0];
  float y1 = ya + ha * d[1];
  float w = wa * expf(d[2]);
  float h = ha * expf(d[3]);
  x1 -= w * 0.5f;
  y1 -= h * 0.5f;
  float x2 = x1 + w, y2 = y1 + h;
  x1 = fminf(fmaxf(x1, 0.0f), IMG);
  y1 = fminf(fmaxf(y1, 0.0f), IMG);
  x2 = fminf(fmaxf(x2, 0.0f), IMG);
  y2 = fminf(fmaxf(y2, 0.0f), IMG);
  boxes[j] = make_float4(x1, y1, x2, y2);
  float sc = scoresRaw[a];
  nsc[j] = 1.0f / (1.0f + expf(-sc));
}

// =====================================================================
// Kernel 6: greedy NMS, 6 picks, IoU>0.5, first-index tie-break; emit rois
// =====================================================================
__global__ __launch_bounds__(1024) void nms_rois(const float4* __restrict__ boxes,
                                                 float* __restrict__ sc,
                                                 const int* __restrict__ counter,
                                                 float* __restrict__ rois) {
  __shared__ float sv[1024];
  __shared__ int si[1024];
  __shared__ float4 sb;
  __shared__ float sarea;
  __shared__ int ssel;
  const int tid = threadIdx.x;
  const int N = counter[0];
  for (int it = 0; it < 6; ++it) {
    float best = -INFINITY;
    int bi = 0x7fffffff;
    for (int i = tid; i < N; i += 1024) {
      float s = sc[i];
      if (s > best || (s == best && i < bi)) { best = s; bi = i; }
    }
    sv[tid] = best; si[tid] = bi;
    __syncthreads();
    for (int off = 512; off > 0; off >>= 1) {
      if (tid < off) {
        if (sv[tid + off] > sv[tid] || (sv[tid + off] == sv[tid] && si[tid + off] < si[tid])) {
          sv[tid] = sv[tid + off];
          si[tid] = si[tid + off];
        }
      }
      __syncthreads();
    }
    if (tid == 0) {
      int s = si[0];
      if (s == 0x7fffffff) s = 0;
      ssel = s;
      float4 b = boxes[s];
      sb = b;
      sarea = (b.z - b.x) * (b.w - b.y);
      rois[it * 4 + 0] = (b.x + b.z) * 0.5f;
      rois[it * 4 + 1] = (b.y + b.w) * 0.5f;
      rois[it * 4 + 2] = b.z - b.x;
      rois[it * 4 + 3] = b.w - b.y;
    }
    __syncthreads();
    float4 b = sb;
    float ab = sarea;
    int sel = ssel;
    for (int i = tid; i < N; i += 1024) {
      float4 c = boxes[i];
      float xx1 = fmaxf(b.x, c.x), yy1 = fmaxf(b.y, c.y);
      float xx2 = fminf(b.z, c.z), yy2 = fminf(b.w, c.w);
      float inter = fmaxf(xx2 - xx1, 0.0f) * fmaxf(yy2 - yy1, 0.0f);
      float area = (c.z - c.x) * (c.w - c.y);
      float iou = inter / (ab + area - inter);
      if (iou > 0.5f || i == sel) sc[i] = NEGF;
    }
    __syncthreads();
  }
}

// =====================================================================
extern "C" void kernel_launch(void* const* d_in, const int* in_sizes, int n_in,
                              void* d_out, int out_size, void* d_ws, size_t ws_size,
                              hipStream_t stream) {
  const float* features = (const float*)d_in[0];
  // d_in[1] = image (only its static shape matters)
  const float* conv1_w = (const float*)d_in[2];
  const float* conv1_b = (const float*)d_in[3];
  const float* reg_w   = (const float*)d_in[4];
  const float* reg_b   = (const float*)d_in[5];
  const float* cls_w   = (const float*)d_in[6];
  const float* cls_b   = (const float*)d_in[7];
  const float* lm_w    = (const float*)d_in[8];
  const float* lm_b    = (const float*)d_in[9];

  float* out = (float*)d_out;
  float* ws  = (float*)d_ws;

  float*  X         = ws + OFF_X;
  float*  Wc        = ws + OFF_WCAT;
  float*  bc        = ws + OFF_BCAT;
  float*  deltas    = ws + OFF_DELTAS;
  float*  scoresRaw = ws + OFF_SCORES;
  float4* anchorsA  = (float4*)(ws + OFF_ANCH);
  float4* boxes     = (float4*)(ws + OFF_BOXES);
  float*  nsc       = ws + OFF_NSC;
  int*    flags     = (int*)(ws + OFF_FLAGS);
  int*    idx       = (int*)(ws + OFF_IDX);
  int*    counter   = (int*)(ws + OFF_CNT);

  float*  roisOut   = out;                      // [6,4]
  float*  lmOut     = out + 24;                 // [16384,24]
  float4* anchOut   = (float4*)(out + 24 + (size_t)HW * 24);  // [N,4]

  pack_weights<<<(512 * NCAT + 255) / 256, 256, 0, stream>>>(
      reg_w, cls_w, lm_w, reg_b, cls_b, lm_b, Wc, bc);
  conv3x3_wmma<<<256, 256, 0, stream>>>(features, conv1_w, conv1_b, X);
  gemm_heads<<<64, 256, 0, stream>>>(X, Wc, bc, deltas, scoresRaw, lmOut);
  gen_anchors<<<ATOT / 256, 256, 0, stream>>>(anchorsA, flags);
  compact_anchors<<<1, 1024, 0, stream>>>(anchorsA, flags, idx, anchOut, counter);
  decode_boxes<<<ATOT / 256, 256, 0, stream>>>(anchorsA, idx, counter, deltas, scoresRaw,
                                               boxes, nsc);
  nms_rois<<<1, 1024, 0, stream>>>(boxes, nsc, counter, roisOut);
}